// crazyleg_5351529251331
// MI455X (gfx1250) — compile-verified
//
#include <hip/hip_runtime.h>
#include <math.h>

#ifndef M_PI
#define M_PI 3.14159265358979323846
#endif

namespace {
// All coefficients computed in double (as Python does) then rounded to f32,
// matching JAX weak-typed scalar -> f32 promotion.
constexpr float CG1  = (float)(-(9.1 * 0.15 + (3.6 + 1.5) * 0.39) * 9.81);
constexpr float CG2  = (float)(-(3.6 * 0.2 + 1.5 * 0.43) * 9.81);
constexpr float KC   = (float)(3.6 * 0.39 * 0.2 + 1.5 * 0.39 * 0.43);
constexpr float M11C = (float)(0.112007 + 9.1 * 0.15 * 0.15 + (3.6 + 1.5) * 0.39 * 0.39);
constexpr float M22C = (float)(0.0591636 + 3.6 * 0.2 * 0.2 + 1.5 * 0.43 * 0.43);
constexpr float HPI  = (float)(M_PI * 0.5);
constexpr float R100 = (float)(100.0 * M_PI / 180.0);
constexpr float R140 = (float)(140.0 * M_PI / 180.0);
constexpr float R90  = (float)( 90.0 * M_PI / 180.0);
constexpr float R50  = (float)( 50.0 * M_PI / 180.0);
// cos/sin of the unit-coefficient phase offsets (double-accurate literals)
// used via cos(q - a) = cos(q)cos(a) + sin(q)sin(a).
constexpr float C65  = 0.42261826174069944f,  S65  = 0.9063077870366499f;
constexpr float C127 = -0.6018150231520483f,  S127 = 0.7986355100472928f;
constexpr float C122 = -0.5299192642332049f,  S122 = 0.8480480961564260f;
constexpr float C85  = 0.08715574274765817f,  S85  = 0.9961946980917455f;
constexpr float SQC  = (float)(0.39 * 0.39 + 0.03 * 0.03);   // LFEM^2 + HSUP^2
constexpr float TWOLH = (float)(2.0 * 0.39 * 0.03);          // 2*LFEM*HSUP
constexpr float RK   = 0.03f;                                // RKNEE
} // namespace

// blend(L) = L*s + (1-s)*0.05 with s = 0.95*sigmoid(100*(L-0.07))
// Branchless, numerically stable logistic with a single expf.
__device__ __forceinline__ float blendf(float Lg) {
  float x = 100.0f * (Lg - 0.07f);
  float e = __builtin_expf(-fabsf(x));
  float inv = 1.0f / (1.0f + e);
  float sg = (x >= 0.0f) ? inv : e * inv;
  float s = 0.95f * sg;
  return Lg * s + (1.0f - s) * 0.05f;
}

__global__ __launch_bounds__(256)
void leg_dyn_kernel(const float* __restrict__ q, const float* __restrict__ qd,
                    float* __restrict__ out, int B) {
  const int stride = (int)(gridDim.x * blockDim.x);
  float* __restrict__ Gq = out;
  float* __restrict__ Cm = out + 2 * (size_t)B;
  float* __restrict__ Mm = out + 6 * (size_t)B;
  float* __restrict__ Rm = out + 10 * (size_t)B;
  float* __restrict__ Lo = out + 24 * (size_t)B;
  const float BASE = sqrtf(SQC);  // constant-folded

  for (int i = (int)(blockIdx.x * blockDim.x + threadIdx.x); i < B; i += stride) {
    // gfx1250 global_prefetch_b8 on the next grid-stride iteration's inputs
    int ip = i + stride;
    if (ip < B) {
      __builtin_prefetch(q + 2 * (size_t)ip, 0, 0);
      __builtin_prefetch(qd + 2 * (size_t)ip, 0, 0);
    }

    float2 qv  = *reinterpret_cast<const float2*>(q  + 2 * (size_t)i);  // b64
    float2 qdv = *reinterpret_cast<const float2*>(qd + 2 * (size_t)i);  // b64
    float q0 = qv.x, q1 = qv.y;
    float qd0 = qdv.x, qd1 = qdv.y;

    // Three sincos evaluations cover g1/g2, M/C, LRF, and 4 R-matrix terms.
    float s0, c0, s1, c1, s01, c01;
    sincosf(q0, &s0, &c0);
    sincosf(q1, &s1, &c1);
    sincosf(q0 - q1, &s01, &c01);

    // ---- Gq ----
    float2 g;
    g.x = CG1 * s0;
    g.y = CG2 * s1;
    *reinterpret_cast<float2*>(Gq + 2 * (size_t)i) = g;  // b64

    // ---- Cmat / Mmat ----
    float ks = KC * s01;
    float4 cm;
    cm.x = 0.0f;
    cm.y = -ks * qd1;   // c12
    cm.z =  ks * qd0;   // c21
    cm.w = 0.0f;
    *reinterpret_cast<float4*>(Cm + 4 * (size_t)i) = cm;  // b128

    float m12 = KC * c01;
    float4 mm;
    mm.x = M11C; mm.y = m12; mm.z = m12; mm.w = M22C;
    *reinterpret_cast<float4*>(Mm + 4 * (size_t)i) = mm;  // b128

    // ---- Rmat (2x7, row-major: 14 floats) ----
    // Unit-coefficient arguments via angle addition (uses s0/c0, s1/c1):
    float RBF   = -0.038f * (c1 * C65  + s1 * S65);   // cos(q1 - 65deg)
    float RRFh  =  0.053f * (c0 * C127 + s0 * S127);  // cos(q0 - 127deg)
    float RSTh  = -0.07f  * (c0 * C122 + s0 * S122);  // cos(q0 - 122deg)
    float RSTk  = -0.04f  * (c1 * C85  + s1 * S85);   // cos(q1 - 85deg)
    // Scaled arguments need their own evaluation:
    float RVlat =  0.05f  * cosf(0.8f  * (q1 - R100));
    float RRFk  =  0.053f * cosf(0.55f * (q1 - R100));
    float RIl   =  0.045f * cosf(0.6f  * (q0 - R140));
    float RGM   = -0.06f  * cosf(0.75f * (q0 - R90));
    float RMG   = -0.038f * cosf(0.7f  * (q1 - R50));

    float2* Rp = reinterpret_cast<float2*>(Rm + 14 * (size_t)i);  // 56B stride: 8-aligned
    Rp[0] = make_float2(RIl,  RGM);
    Rp[1] = make_float2(RRFh, RSTh);
    Rp[2] = make_float2(0.0f, 0.0f);
    Rp[3] = make_float2(0.0f, 0.0f);
    Rp[4] = make_float2(0.0f, RRFk);
    Rp[5] = make_float2(RSTk, RVlat);
    Rp[6] = make_float2(RBF,  RMG);

    // ---- L (7 muscle lengths through blend) ----
    float q0h = q0 - HPI;
    float dq  = q1 - q0;
    float LIl = blendf(0.094f - 0.035f * q0h);
    float LGM = blendf(0.127f + 0.04f  * q0h);
    float LRF = blendf(0.06f + sqrtf(SQC + TWOLH * c0) - BASE - RK * dq);
    float LST = blendf(0.055f + 0.05f * q0h + RK * dq);
    float Lvl = blendf(0.046f + RK * (q0 - q1));
    float Lbf = blendf(0.139f + RK * dq);
    float LMG = blendf(0.055f + RK * dq);

    float* Lp = Lo + 7 * (size_t)i;  // 28B stride: 4-aligned -> scalar (still coalesced)
    Lp[0] = LIl; Lp[1] = LGM; Lp[2] = LRF; Lp[3] = LST;
    Lp[4] = Lvl; Lp[5] = Lbf; Lp[6] = LMG;
  }
}

// Adot = (U - A) / TACT1 over the flat 7B-element arrays: pure streaming,
// float4 (b128) loads/stores, fully aligned (base pointers are hipMalloc'd).
__global__ __launch_bounds__(256)
void adot_kernel(const float* __restrict__ A, const float* __restrict__ U,
                 float* __restrict__ Ad, int n) {
  int n4 = n >> 2;
  int j = (int)(blockIdx.x * blockDim.x + threadIdx.x);
  if (j < n4) {
    float4 a = reinterpret_cast<const float4*>(A)[j];
    float4 u = reinterpret_cast<const float4*>(U)[j];
    float4 d;
    d.x = (u.x - a.x) / 0.12f;
    d.y = (u.y - a.y) / 0.12f;
    d.z = (u.z - a.z) / 0.12f;
    d.w = (u.w - a.w) / 0.12f;
    reinterpret_cast<float4*>(Ad)[j] = d;
  } else if (j == n4) {
    int rem = n & 3;
    int base = n4 << 2;
    for (int k = 0; k < rem; ++k) {
      Ad[base + k] = (U[base + k] - A[base + k]) / 0.12f;
    }
  }
}

extern "C" void kernel_launch(void* const* d_in, const int* in_sizes, int n_in,
                              void* d_out, int out_size, void* d_ws, size_t ws_size,
                              hipStream_t stream) {
  const float* q  = (const float*)d_in[0];
  const float* qd = (const float*)d_in[1];
  const float* A  = (const float*)d_in[2];
  const float* U  = (const float*)d_in[3];
  float* out = (float*)d_out;

  const int B  = in_sizes[0] / 2;   // q is (B, 2)
  const int nA = in_sizes[2];       // A is (B, 7) -> 7B elements

  const int TPB = 256;              // 8 wave32 per block

  // ~2 elements per thread so the grid-stride prefetch actually executes,
  // while still launching enough waves to saturate the WGPs.
  int blocks = (B + 2 * TPB - 1) / (2 * TPB);
  if (blocks < 1) blocks = 1;
  leg_dyn_kernel<<<blocks, TPB, 0, stream>>>(q, qd, out, B);

  int n4 = nA >> 2;
  int ablocks = (n4 + 1 + TPB - 1) / TPB;  // +1 thread covers the tail
  if (ablocks < 1) ablocks = 1;
  adot_kernel<<<ablocks, TPB, 0, stream>>>(A, U, out + 31 * (size_t)B, nA);
}